// ECELoss_6674379178673
// MI455X (gfx1250) — compile-verified
//
#include <hip/hip_runtime.h>
#include <hip/hip_bf16.h>

#define NBINS 15
#define NCLS  256          // classes per row (reference: 256)
#define WPB   8            // waves per block (256 threads, wave32)
#define TPB   256
#define NBUF  4            // LDS ring buffers per wave
#define NPIPE 3            // rows kept in flight (2 async ops each)
#define LOG2E 1.4426950408889634f

typedef __attribute__((address_space(3))) float lds_f32;

// LDS byte-offset of a __shared__-derived pointer (generic -> AS3 addrspacecast).
__device__ __forceinline__ unsigned lds_off(const float* p) {
    return (unsigned)(unsigned long long)(lds_f32*)p;
}

template <int N>
__device__ __forceinline__ void wait_async_le() {
#if defined(__AMDGCN__) && __has_builtin(__builtin_amdgcn_s_wait_asynccnt)
    __builtin_amdgcn_s_wait_asynccnt(N);
    asm volatile("" ::: "memory");
#else
    asm volatile("s_wait_asynccnt %0" :: "i"(N) : "memory");
#endif
}

__device__ __forceinline__ float fast_exp2(float x) {
#if defined(__AMDGCN__) && __has_builtin(__builtin_amdgcn_exp2f)
    return __builtin_amdgcn_exp2f(x);
#else
    return exp2f(x);   // HIP device overload on device pass; glibc decl on host pass
#endif
}

// Async-DMA one 256-float row (1024 B) into an LDS buffer, non-temporal
// (1 GB stream touched once: don't churn the 192 MB L2).
// Each of the 32 lanes moves 16 B at byte l*16 and, via the instruction
// offset (added to BOTH LDS and global addresses), at l*16+512.
__device__ __forceinline__ void async_load_row(const float* gsrc, float* lbuf, int lane) {
    unsigned           laddr = lds_off(lbuf + lane * 4);
    unsigned long long gaddr = (unsigned long long)(gsrc + lane * 4);
    asm volatile(
        "global_load_async_to_lds_b128 %0, %1, off th:TH_LOAD_NT\n\t"
        "global_load_async_to_lds_b128 %0, %1, off offset:512 th:TH_LOAD_NT"
        :: "v"(laddr), "v"(gaddr)
        : "memory");
}

__global__ void ece_init_ws(float* ws) {
    if (threadIdx.x < 3 * NBINS) ws[threadIdx.x] = 0.0f;
}

__global__ __launch_bounds__(TPB) void ece_main(const float* __restrict__ logits,
                                                const long long* __restrict__ labels,
                                                float* __restrict__ ws, int nrows) {
    __shared__ float buf[WPB][NBUF][NCLS];   // 32 KB: 4-deep ring per wave
    __shared__ float sb[3 * NBINS];

    const int tid  = threadIdx.x;
    const int lane = tid & 31;
    const int wv   = tid >> 5;

    if (tid < 3 * NBINS) sb[tid] = 0.0f;
    __syncthreads();

    const int    gwave   = blockIdx.x * WPB + wv;
    const int    stride  = gridDim.x * WPB;
    const size_t rstride = (size_t)stride * NCLS;

    float cnt = 0.0f, csum = 0.0f, asum = 0.0f;

    // ---- prologue: fill the pipeline with up to NPIPE rows ----
    int          inflight = 0;
    int          irow     = gwave;                      // next row to issue
    const float* isrc     = logits + (size_t)gwave * NCLS;
    int          ibuf     = 0;
    #pragma unroll
    for (int k = 0; k < NPIPE; ++k) {
        if (irow < nrows) {
            async_load_row(isrc, buf[wv][ibuf], lane);
            irow += stride; isrc += rstride;
            ibuf = (ibuf + 1) & (NBUF - 1);
            ++inflight;
        }
    }

    int cbuf = 0;
    for (int row = gwave; row < nrows; row += stride) {
        const int lab = (int)labels[row];               // overlap with LDS reads

        // oldest outstanding row done when asynccnt <= 2*(inflight-1)
        // (async loads complete in order; 2 ops per row)
        if      (inflight >= 3) wait_async_le<4>();
        else if (inflight == 2) wait_async_le<2>();
        else                    wait_async_le<0>();

        if (irow < nrows) {                              // keep pipeline full
            async_load_row(isrc, buf[wv][ibuf], lane);
            irow += stride; isrc += rstride;
            ibuf = (ibuf + 1) & (NBUF - 1);
        } else {
            --inflight;                                  // drain tail
        }

        const float4* rb4 = (const float4*)buf[wv][cbuf];
        const float4  a   = rb4[lane];                   // elems lane*4 .. +3
        const float4  b   = rb4[32 + lane];              // elems 128+lane*4 .. +3
        const float   e[8] = {a.x, a.y, a.z, a.w, b.x, b.y, b.z, b.w};

        // ---- global max: 7-op local tree + 5-step max butterfly ----
        float m = fmaxf(fmaxf(fmaxf(e[0], e[1]), fmaxf(e[2], e[3])),
                        fmaxf(fmaxf(e[4], e[5]), fmaxf(e[6], e[7])));
        #pragma unroll
        for (int off = 16; off >= 1; off >>= 1)
            m = fmaxf(m, __shfl_xor(m, off));

        // ---- first index attaining max (argmax, first-occurrence) ----
        // descending scan: lowest matching local index survives
        const int base = lane * 4;
        int cand = 0x7FFFFFFF;
        #pragma unroll
        for (int k = 7; k >= 0; --k) {
            const int i = (k < 4) ? (base + k) : (124 + base + k);  // 128+base+(k-4)
            cand = (e[k] == m) ? i : cand;
        }
        #pragma unroll
        for (int off = 16; off >= 1; off >>= 1)
            cand = min(cand, __shfl_xor(cand, off));

        // ---- sum of exp(x - max) via fused fma + native exp2 ----
        const float m2 = m * LOG2E;
        float s0 = 0.0f, s1 = 0.0f;
        #pragma unroll
        for (int k = 0; k < 4; ++k) {
            s0 += fast_exp2(__builtin_fmaf(e[k],     LOG2E, -m2));
            s1 += fast_exp2(__builtin_fmaf(e[k + 4], LOG2E, -m2));
        }
        float s = s0 + s1;
        #pragma unroll
        for (int off = 16; off >= 1; off >>= 1) s += __shfl_xor(s, off);

        const float conf = 1.0f / s;
        const float accf = (cand == lab) ? 1.0f : 0.0f;

        // bin i covers (i/15, (i+1)/15] -> ceil(conf*15)-1, clipped
        int bn = (int)ceilf(conf * (float)NBINS) - 1;
        bn = bn < 0 ? 0 : (bn > NBINS - 1 ? NBINS - 1 : bn);

        if (lane == bn) { cnt += 1.0f; csum += conf; asum += accf; }
        cbuf = (cbuf + 1) & (NBUF - 1);
    }

    // lane b holds this wave's bin-b partials -> block LDS -> global
    if (lane < NBINS) {
        atomicAdd(&sb[lane],             cnt);
        atomicAdd(&sb[NBINS + lane],     csum);
        atomicAdd(&sb[2 * NBINS + lane], asum);
    }
    __syncthreads();
    if (tid < 3 * NBINS) {
        float v = sb[tid];
        if (v != 0.0f) atomicAdd(&ws[tid], v);
    }
}

__global__ void ece_finalize(const float* __restrict__ ws, float* __restrict__ out,
                             float inv_n) {
    const int lane = threadIdx.x;
    float term = 0.0f;
    if (lane < NBINS) {
        float c = ws[lane];
        if (c > 0.0f) {
            float avg_conf = ws[NBINS + lane]     / c;
            float avg_acc  = ws[2 * NBINS + lane] / c;
            term = fabsf(avg_conf - avg_acc) * c * inv_n;
        }
    }
    #pragma unroll
    for (int off = 16; off >= 1; off >>= 1) term += __shfl_xor(term, off);
    if (lane == 0) out[0] = term;
}

extern "C" void kernel_launch(void* const* d_in, const int* in_sizes, int n_in,
                              void* d_out, int out_size, void* d_ws, size_t ws_size,
                              hipStream_t stream) {
    const float*     logits = (const float*)d_in[0];
    const long long* labels = (const long long*)d_in[1];   // jnp.int64
    float*           ws     = (float*)d_ws;                // 45 floats
    float*           out    = (float*)d_out;

    const int nrows = in_sizes[1];

    int blocks = (nrows + WPB - 1) / WPB;
    if (blocks > 2048) blocks = 2048;   // 16K waves * 3KB in flight ~= 48MB in flight
    if (blocks < 1)    blocks = 1;

    ece_init_ws<<<1, 64, 0, stream>>>(ws);
    ece_main<<<blocks, TPB, 0, stream>>>(logits, labels, ws, nrows);
    ece_finalize<<<1, 32, 0, stream>>>(ws, out, 1.0f / (float)nrows);
}